// Dynamics_10857677324868
// MI455X (gfx1250) — compile-verified
//
#include <hip/hip_runtime.h>
#include <cstddef>

typedef float v2f __attribute__((ext_vector_type(2)));
typedef float v8f __attribute__((ext_vector_type(8)));

#define B_TOT   524288
#define NTILES  (B_TOT / 16)   // 32768 tiles of 16 samples
#define WPB     4              // waves per block (wave32)
#define THREADS (WPB * 32)
#define NBLOCKS 2048
#define EPSC    0.01f

union U8 { v8f v; float f[8]; };

__device__ __forceinline__ v8f wmma4(v2f a, v2f b, v8f c) {
  // D = A(16x4,f32) * B(4x16,f32) + C(16x16,f32)
  return __builtin_amdgcn_wmma_f32_16x16x4_f32(false, a, false, b, (short)0, c, false, false);
}

__device__ __forceinline__ float rehu(float z, float d) {
  // max(clamp(sign(z)*z^2/(2d), 0, d/2), z - d/2)
  float q  = (0.5f / d) * z * fabsf(z);
  float cl = fminf(fmaxf(q, 0.0f), 0.5f * d);
  return fmaxf(cl, z - 0.5f * d);
}
__device__ __forceinline__ float clamp01(float z) { return fminf(fmaxf(z, 0.0f), 1.0f); }

// Forward MLP trunk for one 16-sample tile given layer-1 A-fragments.
// Writes ReLU(H1) into buf (16x48), returns ReLU(H2) tiles in h2[3] (C/D layout).
__device__ __forceinline__ void forward_net(const v2f a1frag[2], float* buf,
                                            const float* sW1T, const float* sW2T,
                                            const float c1v[3], const float c2v[3],
                                            int lm, int half, U8 h2[3]) {
  // ---- layer 1: [16,8] x [8,40->48] ----
#pragma unroll
  for (int nt = 0; nt < 3; ++nt) {
    v8f acc = {0.f,0.f,0.f,0.f,0.f,0.f,0.f,0.f};
#pragma unroll
    for (int kk = 0; kk < 2; ++kk) {
      int k0 = 4 * kk + 2 * half;
      v2f b = *(const v2f*)(sW1T + (lm + 16 * nt) * 8 + k0);
      acc = wmma4(a1frag[kk], b, acc);
    }
    U8 u; u.v = acc;
    float bias = c1v[nt];
#pragma unroll
    for (int i = 0; i < 8; ++i) {
      float h = fmaxf(u.f[i] + bias, 0.0f);
      buf[(i + 8 * half) * 48 + lm + 16 * nt] = h;   // H1[m][n]
    }
  }
  // ---- layer 2: [16,40] x [40,40->48] ----
  v2f af[10];
#pragma unroll
  for (int kk = 0; kk < 10; ++kk)
    af[kk] = *(const v2f*)(buf + lm * 48 + 4 * kk + 2 * half);  // A-frag H1
#pragma unroll
  for (int nt = 0; nt < 3; ++nt) {
    v8f acc = {0.f,0.f,0.f,0.f,0.f,0.f,0.f,0.f};
#pragma unroll
    for (int kk = 0; kk < 10; ++kk) {
      int k0 = 4 * kk + 2 * half;
      v2f b = *(const v2f*)(sW2T + (lm + 16 * nt) * 40 + k0);
      acc = wmma4(af[kk], b, acc);
    }
    U8 u; u.v = acc;
    float bias = c2v[nt];
#pragma unroll
    for (int i = 0; i < 8; ++i) u.f[i] = fmaxf(u.f[i] + bias, 0.0f);
    h2[nt] = u;
  }
}

__global__ __launch_bounds__(THREADS)
void lyap_kernel(const float* __restrict__ x,  const float* __restrict__ g,
                 const float* __restrict__ A1, const float* __restrict__ c1,
                 const float* __restrict__ A2, const float* __restrict__ c2,
                 const float* __restrict__ A3, const float* __restrict__ c3,
                 float* __restrict__ blockSums) {
  // Padded weight images in LDS (zero-padded so WMMA never needs predication)
  __shared__ float sW1T[48 * 8];    // [n][k] = A1[n][k]  (B for layer1: A1^T)
  __shared__ float sW2T[48 * 40];   // [n][k] = A2[n][k]  (B for layer2: A2^T)
  __shared__ float sW2P[40 * 48];   // [k][n] = A2[k][n]  (B for bwd layer2)
  __shared__ float sW1P[40 * 16];   // [k][n] = A1[k][n]  (B for bwd layer1)
  __shared__ float sA3[48], sC1[48], sC2[48];
  __shared__ float sMisc[2];        // [0]=c3, [1]=f0
  __shared__ float sBuf1[WPB][16 * 48];
  __shared__ float sBuf2[WPB][16 * 48];
  __shared__ float sWaveSum[WPB];

  const int tid = threadIdx.x;
  for (int i = tid; i < 48 * 8;  i += THREADS) { int n = i / 8,  k = i % 8;  sW1T[i] = (n < 40) ? A1[n * 8 + k]  : 0.f; }
  for (int i = tid; i < 48 * 40; i += THREADS) { int n = i / 40, k = i % 40; sW2T[i] = (n < 40) ? A2[n * 40 + k] : 0.f; }
  for (int i = tid; i < 40 * 48; i += THREADS) { int k = i / 48, n = i % 48; sW2P[i] = (n < 40) ? A2[k * 40 + n] : 0.f; }
  for (int i = tid; i < 40 * 16; i += THREADS) { int k = i / 16, n = i % 16; sW1P[i] = (n < 8)  ? A1[k * 8 + n]  : 0.f; }
  for (int i = tid; i < 48; i += THREADS) {
    sA3[i] = (i < 40) ? A3[i] : 0.f;
    sC1[i] = (i < 40) ? c1[i] : 0.f;
    sC2[i] = (i < 40) ? c2[i] : 0.f;
  }
  if (tid == 0) {
    sMisc[0] = c3[0];
    float h1l[40];
    for (int n = 0; n < 40; ++n) h1l[n] = fmaxf(c1[n], 0.f);
    float y0 = c3[0];
    for (int n = 0; n < 40; ++n) {
      float acc = c2[n];
      for (int k = 0; k < 40; ++k) acc += A2[n * 40 + k] * h1l[k];
      y0 += A3[n] * fmaxf(acc, 0.f);
    }
    sMisc[1] = rehu(y0, 0.01f);   // f(0)
  }
  __syncthreads();

  const int wid  = tid >> 5;
  const int lane = tid & 31;
  const int half = lane >> 4;     // which 16-lane half
  const int lm   = lane & 15;
  float* buf1 = sBuf1[wid];
  float* buf2 = sBuf2[wid];
  const float f0  = sMisc[1];
  const float c3v = sMisc[0];
  float a3v[3], c1v[3], c2v[3];
#pragma unroll
  for (int nt = 0; nt < 3; ++nt) {
    a3v[nt] = sA3[lm + 16 * nt];
    c1v[nt] = sC1[lm + 16 * nt];
    c2v[nt] = sC2[lm + 16 * nt];
  }

  float riskA = 0.f, riskB = 0.f, riskC = 0.f;
  const int gwave  = blockIdx.x * WPB + wid;
  const int nWaves = gridDim.x * WPB;

  for (int tile = gwave; tile < NTILES; tile += nWaves) {
    const int sbase = tile * 16;

    // ================= forward pass on x =================
    v2f ax[2]; float sx2p = 0.f;
#pragma unroll
    for (int kk = 0; kk < 2; ++kk) {
      int k0 = 4 * kk + 2 * half;
      v2f t = *(const v2f*)(x + (size_t)(sbase + lm) * 8 + k0);
      ax[kk] = t; sx2p += t.x * t.x + t.y * t.y;
    }
    float sx2 = sx2p + __shfl_xor(sx2p, 16, 32);   // ||x_m||^2 at lanes m, m+16

    U8 h2[3];
    forward_net(ax, buf1, sW1T, sW2T, c1v, c2v, lm, half, h2);

    // layer 3 + per-sample scalars (y replicated within each 16-lane half)
    float sv[8];
#pragma unroll
    for (int i = 0; i < 8; ++i) {
      float a = h2[0].f[i] * a3v[0] + h2[1].f[i] * a3v[1] + h2[2].f[i] * a3v[2];
      a += __shfl_xor(a, 1, 32); a += __shfl_xor(a, 2, 32);
      a += __shfl_xor(a, 4, 32); a += __shfl_xor(a, 8, 32);
      float y = a + c3v;                      // y for sample m = i + 8*half
      float f = rehu(y, 0.01f);
      float z = f - f0;
      sv[i] = clamp01(z) * clamp01(y * 100.0f);   // combined rehu' chain
      float sx2m = __shfl(sx2, i + 8 * half, 32);
      float Vx   = rehu(z, 1.0f) + EPSC * sx2m;
      riskA += (lm == i) ? fmaxf(-Vx, 0.f) : 0.f;  // relu(-Vx) (=0, kept for fidelity)
    }

    // v2 = s * A3 .* relu'(H2pre)  -> buf2
#pragma unroll
    for (int nt = 0; nt < 3; ++nt)
#pragma unroll
      for (int i = 0; i < 8; ++i)
        buf2[(i + 8 * half) * 48 + lm + 16 * nt] =
            (h2[nt].f[i] > 0.f) ? sv[i] * a3v[nt] : 0.f;

    // ================= backward GEMM2: u1 = v2 @ A2 =================
    {
      v2f af[10];
#pragma unroll
      for (int kk = 0; kk < 10; ++kk)
        af[kk] = *(const v2f*)(buf2 + lm * 48 + 4 * kk + 2 * half);
#pragma unroll
      for (int nt = 0; nt < 3; ++nt) {
        v8f acc = {0.f,0.f,0.f,0.f,0.f,0.f,0.f,0.f};
#pragma unroll
        for (int kk = 0; kk < 10; ++kk) {
          int k0 = 4 * kk + 2 * half;
          int n  = lm + 16 * nt;
          v2f b; b.x = sW2P[k0 * 48 + n]; b.y = sW2P[(k0 + 1) * 48 + n];
          acc = wmma4(af[kk], b, acc);
        }
        U8 u; u.v = acc;
#pragma unroll
        for (int i = 0; i < 8; ++i) {
          int m = i + 8 * half, n = lm + 16 * nt;
          float h1v = buf1[m * 48 + n];
          buf2[m * 48 + n] = (h1v > 0.f) ? u.f[i] : 0.f;   // v1
        }
      }
    }

    // ================= backward GEMM1: gx = v1 @ A1 =================
    {
      v2f af[10];
#pragma unroll
      for (int kk = 0; kk < 10; ++kk)
        af[kk] = *(const v2f*)(buf2 + lm * 48 + 4 * kk + 2 * half);
      v8f acc = {0.f,0.f,0.f,0.f,0.f,0.f,0.f,0.f};
#pragma unroll
      for (int kk = 0; kk < 10; ++kk) {
        int k0 = 4 * kk + 2 * half;
        v2f b; b.x = sW1P[k0 * 16 + lm]; b.y = sW1P[(k0 + 1) * 16 + lm];
        acc = wmma4(af[kk], b, acc);
      }
      U8 u; u.v = acc;
      int nsafe = (lm < 8) ? lm : 0;
#pragma unroll
      for (int i = 0; i < 8; ++i) {
        int m = i + 8 * half;
        float xv = x[(size_t)(sbase + m) * 8 + nsafe];
        float gv = u.f[i] + 2.f * EPSC * xv;
        riskB += (lm < 8) ? fmaxf(gv, 0.f) : 0.f;   // sum relu(gV)
      }
    }

    // ================= forward pass on x0 = |x - g| =================
    v2f a0[2]; float s0p = 0.f;
#pragma unroll
    for (int kk = 0; kk < 2; ++kk) {
      int k0 = 4 * kk + 2 * half;
      size_t idx = (size_t)(sbase + lm) * 8 + k0;
      v2f xv = *(const v2f*)(x + idx);
      v2f gv = *(const v2f*)(g + idx);
      v2f a; a.x = fabsf(xv.x - gv.x); a.y = fabsf(xv.y - gv.y);
      a0[kk] = a; s0p += a.x * a.x + a.y * a.y;
    }
    float sx0 = s0p + __shfl_xor(s0p, 16, 32);

    U8 h2b[3];
    forward_net(a0, buf1, sW1T, sW2T, c1v, c2v, lm, half, h2b);

#pragma unroll
    for (int i = 0; i < 8; ++i) {
      float a = h2b[0].f[i] * a3v[0] + h2b[1].f[i] * a3v[1] + h2b[2].f[i] * a3v[2];
      a += __shfl_xor(a, 1, 32); a += __shfl_xor(a, 2, 32);
      a += __shfl_xor(a, 4, 32); a += __shfl_xor(a, 8, 32);
      float y  = a + c3v;
      float f  = rehu(y, 0.01f);
      float sm = __shfl(sx0, i + 8 * half, 32);
      float X0 = rehu(f - f0, 1.0f) + EPSC * sm;
      riskC += (lm == i) ? X0 * X0 : 0.f;
    }
  }

  // risk = riskA/B + riskB/(4B) + 10*riskC, reduced deterministically
  float part = riskA * (1.0f / (float)B_TOT)
             + riskB * (1.0f / (4.0f * (float)B_TOT))
             + riskC * 10.0f;
  part += __shfl_xor(part, 1, 32);  part += __shfl_xor(part, 2, 32);
  part += __shfl_xor(part, 4, 32);  part += __shfl_xor(part, 8, 32);
  part += __shfl_xor(part, 16, 32);
  if (lane == 0) sWaveSum[wid] = part;
  __syncthreads();
  if (tid == 0) {
    float s = 0.f;
    for (int w = 0; w < WPB; ++w) s += sWaveSum[w];
    blockSums[blockIdx.x] = s;
  }
}

__global__ __launch_bounds__(256)
void reduce_kernel(const float* __restrict__ ws, float* __restrict__ out, int n) {
  __shared__ float sh[256];
  int tid = threadIdx.x;
  float acc = 0.f;
  for (int i = tid; i < n; i += 256) acc += ws[i];
  sh[tid] = acc;
  __syncthreads();
  for (int s = 128; s > 0; s >>= 1) {
    if (tid < s) sh[tid] += sh[tid + s];
    __syncthreads();
  }
  if (tid == 0) out[0] = sh[0];
}

extern "C" void kernel_launch(void* const* d_in, const int* in_sizes, int n_in,
                              void* d_out, int out_size, void* d_ws, size_t ws_size,
                              hipStream_t stream) {
  (void)in_sizes; (void)n_in; (void)out_size; (void)ws_size;
  const float* x  = (const float*)d_in[0];
  const float* g  = (const float*)d_in[1];
  // d_in[2..7] = W1,b1,W2,b2,W3,b3 (unused by the reference computation)
  const float* A1 = (const float*)d_in[8];
  const float* c1 = (const float*)d_in[9];
  const float* A2 = (const float*)d_in[10];
  const float* c2 = (const float*)d_in[11];
  const float* A3 = (const float*)d_in[12];
  const float* c3 = (const float*)d_in[13];
  float* ws  = (float*)d_ws;
  float* out = (float*)d_out;

  lyap_kernel<<<NBLOCKS, THREADS, 0, stream>>>(x, g, A1, c1, A2, c2, A3, c3, ws);
  reduce_kernel<<<1, 256, 0, stream>>>(ws, out, NBLOCKS);
}